// Cifp_48206712930739
// MI455X (gfx1250) — compile-verified
//
#include <hip/hip_runtime.h>
#include <math.h>

#define N_ROWS 256
#define D_DIM  512
#define C_COLS 100000
#define C4     (C_COLS / 4)
#define NC_TOT (N_ROWS * C_COLS)
#define SCALE_F 64.0f
#define INV_SCALE_F (1.0f / 64.0f)
#define MARGIN_F 0.35f

typedef float v2f __attribute__((ext_vector_type(2)));
typedef float v4f __attribute__((ext_vector_type(4)));
typedef float v8f __attribute__((ext_vector_type(8)));

// ---------------- workspace layout (units of 4 bytes) ----------------
#define WS_NORMA 0                      // 256*512 floats
#define WS_TGT   131072                 // 256 floats
#define WS_SCAL  131328                 // 16 u32: [0]=topk_sum [1]=r [2]=hi1 [3]=r1 [4]=sel2 [5]=r2 [6]=th bits
#define WS_H1    131344                 // 4096 u32
#define WS_H2    135440                 // 4096 u32
#define WS_H3    139536                 // 256 u32
#define WS_ZERO_COUNT (16 + 4096 + 4096 + 256)

__device__ __forceinline__ unsigned f2key(float f) {
    unsigned u = __float_as_uint(f);
    return (u & 0x80000000u) ? ~u : (u | 0x80000000u);
}
__device__ __forceinline__ float key2f(unsigned k) {
    unsigned u = (k & 0x80000000u) ? (k & 0x7fffffffu) : ~k;
    return __uint_as_float(u);
}

// ---------------- zero counters/histograms ----------------
__global__ void ws_init_kernel(unsigned* __restrict__ scal_base) {
    for (int i = blockIdx.x * 256 + threadIdx.x; i < WS_ZERO_COUNT; i += gridDim.x * 256)
        scal_base[i] = 0u;
}

// ---------------- row-normalize embeddings ----------------
__global__ __launch_bounds__(256) void rownorm_kernel(const float* __restrict__ emb,
                                                      float* __restrict__ normA) {
    const int n = blockIdx.x, t = threadIdx.x;
    float x0 = emb[n * D_DIM + t];
    float x1 = emb[n * D_DIM + 256 + t];
    __shared__ float red[256];
    red[t] = x0 * x0 + x1 * x1;
    __syncthreads();
    for (int o = 128; o > 0; o >>= 1) {
        if (t < o) red[t] += red[t + o];
        __syncthreads();
    }
    __shared__ float invs;
    if (t == 0) invs = 1.0f / fmaxf(sqrtf(red[0]), 1e-12f);
    __syncthreads();
    normA[n * D_DIM + t]       = x0 * invs;
    normA[n * D_DIM + 256 + t] = x1 * invs;
}

// ---------------- WMMA f32 GEMM with fused column norms ----------------
// 8 waves / block; block computes 256 rows x 64 cols.
// wave w: column group (w&3)*16, M-half (w>>2)*128 (8 M-subtiles of 16).
// Column sum-of-squares accumulated in the k-loop (lanes l / l+16 cover
// k = {0,1} / {2,3} mod 4 of the SAME column) -> one shfl_xor(16) finishes it.
__global__ __launch_bounds__(256) void gemm_cos_kernel(const float* __restrict__ A,
                                                       const float* __restrict__ K,
                                                       float* __restrict__ out) {
    __shared__ float As[256 * 36];  // K-chunk of 32, padded stride 36 (conflict-free)
    const int lane  = threadIdx.x & 31;
    const int wid   = threadIdx.x >> 5;
    const int cg    = wid & 3;
    const int mhalf = wid >> 2;
    const int colBase = blockIdx.x * 64;
    const int col  = colBase + cg * 16 + (lane & 15);
    const int colc = (col < C_COLS) ? col : (C_COLS - 1);
    const int khi  = (lane >> 4) * 2;   // 0 for lanes 0-15, 2 for lanes 16-31
    const int rL   = lane & 15;

    v8f acc[8];
#pragma unroll
    for (int m = 0; m < 8; ++m)
#pragma unroll
        for (int v = 0; v < 8; ++v) acc[m][v] = 0.0f;

    float bsq = 0.0f;   // partial column sum-of-squares (this lane's K slices)

    for (int kc = 0; kc < D_DIM / 32; ++kc) {
        __syncthreads();
        // async-stage A chunk (rows 0..255, cols kc*32..+32) into LDS
#pragma unroll
        for (int j = 0; j < 8; ++j) {
            int i   = threadIdx.x + j * 256;
            int row = i >> 3;
            int q   = i & 7;
            const float* gp = &A[row * D_DIM + kc * 32 + q * 4];
            unsigned lds_off = (unsigned)(size_t)&As[row * 36 + q * 4];
            asm volatile("global_load_async_to_lds_b128 %0, %1, off"
                         :: "v"(lds_off), "v"(gp) : "memory");
        }
        asm volatile("s_wait_asynccnt 0" ::: "memory");
        __syncthreads();
#pragma unroll
        for (int kk = 0; kk < 8; ++kk) {
            const int kg = kc * 32 + kk * 4 + khi;
            v2f b;
            b.x = K[(size_t)kg * C_COLS + colc];
            b.y = K[(size_t)(kg + 1) * C_COLS + colc];
            bsq += b.x * b.x + b.y * b.y;
#pragma unroll
            for (int m = 0; m < 8; ++m) {
                const int lrow = mhalf * 128 + m * 16 + rL;
                v2f a = *(const v2f*)&As[lrow * 36 + kk * 4 + khi];
                acc[m] = __builtin_amdgcn_wmma_f32_16x16x4_f32(
                    false, a, false, b, (short)0, acc[m], false, false);
            }
        }
    }

    // finish column norm: combine the two half-wave partial sums
    const float colsq = bsq + __shfl_xor(bsq, 16, 32);
    const float inv   = 1.0f / fmaxf(sqrtf(colsq), 1e-12f);

    if (col < C_COLS) {
#pragma unroll
        for (int m = 0; m < 8; ++m) {
#pragma unroll
            for (int v = 0; v < 8; ++v) {
                const int row = mhalf * 128 + m * 16 + v + 8 * (lane >> 4);
                float cosv = fminf(fmaxf(acc[m][v] * inv, -1.0f), 1.0f);
                float o = cosv * SCALE_F;
                out[(size_t)row * C_COLS + col] = o;                   // cos_out half (fixed later)
                out[(size_t)NC_TOT + (size_t)row * C_COLS + col] = o;  // origin_cos half
            }
        }
    }
}

// ---------------- gather target logits ----------------
__global__ void tgt_kernel(const float* __restrict__ out, const int* __restrict__ label,
                           float* __restrict__ tgt) {
    int n = threadIdx.x;
    if (n < N_ROWS) tgt[n] = out[(size_t)n * C_COLS + label[n]] * INV_SCALE_F;
}

// ---------------- radix-select level 1: count topk + hist of key[31:20] ----------------
// grid: dim3(GX, 256): blockIdx.y = row, b128 loads, no integer division.
__global__ __launch_bounds__(256) void hist1_kernel(const float* __restrict__ out,
                                                    const int* __restrict__ label,
                                                    const float* __restrict__ tgt,
                                                    unsigned* __restrict__ scal,
                                                    unsigned* __restrict__ h1) {
    __shared__ unsigned lh[4096];
    __shared__ unsigned ccount;
    for (int i = threadIdx.x; i < 4096; i += 256) lh[i] = 0u;
    if (threadIdx.x == 0) ccount = 0u;
    __syncthreads();
    const int n = blockIdx.y;
    const int lab = label[n];
    const float tg = tgt[n];
    const float* rowp = out + (size_t)n * C_COLS;
    const int stride = gridDim.x * 256;
    unsigned cnt = 0;
    for (int i = blockIdx.x * 256 + threadIdx.x; i < C4; i += stride) {
        v4f cv = *(const v4f*)&rowp[i * 4];
#pragma unroll
        for (int e = 0; e < 4; ++e) {
            int c = i * 4 + e;
            float tmp = cv[e] * INV_SCALE_F - ((c == lab) ? 2.0f : 0.0f);
            int topk = (tmp > tg) ? 1 : 0;
            cnt += topk;
            atomicAdd(&lh[f2key(tmp - 2.0f * (float)topk) >> 20], 1u);
        }
    }
    atomicAdd(&ccount, cnt);
    __syncthreads();
    for (int i = threadIdx.x; i < 4096; i += 256)
        if (lh[i]) atomicAdd(&h1[i], lh[i]);
    if (threadIdx.x == 0 && ccount) atomicAdd(&scal[0], ccount);
}

__global__ void scan1_kernel(unsigned* __restrict__ scal, const unsigned* __restrict__ h1) {
    if (threadIdx.x | blockIdx.x) return;
    unsigned topk_sum = scal[0];
    double far = 1.0 / (double)(C_COLS - 1);
    long long tot = (long long)N_ROWS * (C_COLS - 1) - (long long)topk_sum;
    long long fr  = (long long)ceil(far * (double)tot);
    long long idx = fr - 1;
    if (idx < 0) idx = 0;
    if (idx > N_ROWS - 1) idx = N_ROWS - 1;
    unsigned r = (unsigned)(idx + 1);
    scal[1] = r;
    unsigned cum = 0;
    for (int b = 4095; b >= 0; --b) {
        unsigned h = h1[b];
        if (cum + h >= r) { scal[2] = (unsigned)b; scal[3] = r - cum; return; }
        cum += h;
    }
}

// ---------------- level 2: hist of key[19:8] among hi-bucket ----------------
__global__ __launch_bounds__(256) void hist2_kernel(const float* __restrict__ out,
                                                    const int* __restrict__ label,
                                                    const float* __restrict__ tgt,
                                                    const unsigned* __restrict__ scal,
                                                    unsigned* __restrict__ h2) {
    __shared__ unsigned lh[4096];
    for (int i = threadIdx.x; i < 4096; i += 256) lh[i] = 0u;
    __syncthreads();
    const unsigned hi1 = scal[2];
    const int n = blockIdx.y;
    const int lab = label[n];
    const float tg = tgt[n];
    const float* rowp = out + (size_t)n * C_COLS;
    const int stride = gridDim.x * 256;
    for (int i = blockIdx.x * 256 + threadIdx.x; i < C4; i += stride) {
        v4f cv = *(const v4f*)&rowp[i * 4];
#pragma unroll
        for (int e = 0; e < 4; ++e) {
            int c = i * 4 + e;
            float tmp = cv[e] * INV_SCALE_F - ((c == lab) ? 2.0f : 0.0f);
            int topk = (tmp > tg) ? 1 : 0;
            unsigned key = f2key(tmp - 2.0f * (float)topk);
            if ((key >> 20) == hi1) atomicAdd(&lh[(key >> 8) & 0xFFFu], 1u);
        }
    }
    __syncthreads();
    for (int i = threadIdx.x; i < 4096; i += 256)
        if (lh[i]) atomicAdd(&h2[i], lh[i]);
}

__global__ void scan2_kernel(unsigned* __restrict__ scal, const unsigned* __restrict__ h2) {
    if (threadIdx.x | blockIdx.x) return;
    unsigned r1 = scal[3], hi1 = scal[2], cum = 0;
    for (int b = 4095; b >= 0; --b) {
        unsigned h = h2[b];
        if (cum + h >= r1) { scal[4] = (hi1 << 12) | (unsigned)b; scal[5] = r1 - cum; return; }
        cum += h;
    }
}

// ---------------- level 3: hist of key[7:0] among 24-bit prefix ----------------
__global__ __launch_bounds__(256) void hist3_kernel(const float* __restrict__ out,
                                                    const int* __restrict__ label,
                                                    const float* __restrict__ tgt,
                                                    const unsigned* __restrict__ scal,
                                                    unsigned* __restrict__ h3) {
    __shared__ unsigned lh[256];
    lh[threadIdx.x] = 0u;
    __syncthreads();
    const unsigned sel2 = scal[4];
    const int n = blockIdx.y;
    const int lab = label[n];
    const float tg = tgt[n];
    const float* rowp = out + (size_t)n * C_COLS;
    const int stride = gridDim.x * 256;
    for (int i = blockIdx.x * 256 + threadIdx.x; i < C4; i += stride) {
        v4f cv = *(const v4f*)&rowp[i * 4];
#pragma unroll
        for (int e = 0; e < 4; ++e) {
            int c = i * 4 + e;
            float tmp = cv[e] * INV_SCALE_F - ((c == lab) ? 2.0f : 0.0f);
            int topk = (tmp > tg) ? 1 : 0;
            unsigned key = f2key(tmp - 2.0f * (float)topk);
            if ((key >> 8) == sel2) atomicAdd(&lh[key & 0xFFu], 1u);
        }
    }
    __syncthreads();
    if (lh[threadIdx.x]) atomicAdd(&h3[threadIdx.x], lh[threadIdx.x]);
}

__global__ void scan3_kernel(unsigned* __restrict__ scal, const unsigned* __restrict__ h3) {
    if (threadIdx.x | blockIdx.x) return;
    unsigned r2 = scal[5], sel2 = scal[4], cum = 0;
    for (int b = 255; b >= 0; --b) {
        unsigned h = h3[b];
        if (cum + h >= r2) {
            scal[6] = __float_as_uint(key2f((sel2 << 8) | (unsigned)b));
            return;
        }
        cum += h;
    }
}

// ---------------- per-row neg-mean + label fix ----------------
__global__ __launch_bounds__(256) void negfix_kernel(float* __restrict__ out,
                                                     const int* __restrict__ label,
                                                     const float* __restrict__ tgt,
                                                     const unsigned* __restrict__ scal) {
    const int n = blockIdx.x, t = threadIdx.x;
    const int lab = label[n];
    const float tg = tgt[n];
    const float th = __uint_as_float(scal[6]);
    const float* rowp = out + (size_t)n * C_COLS;
    float s = 0.0f;
    int cnt = 0;
    for (int i = t; i < C4; i += 256) {
        v4f cv = *(const v4f*)&rowp[i * 4];
#pragma unroll
        for (int e = 0; e < 4; ++e) {
            int c = i * 4 + e;
            float tmp = cv[e] * INV_SCALE_F - ((c == lab) ? 2.0f : 0.0f);
            bool topk = tmp > tg;
            bool cond = (!topk) && (tmp > th);
            float v  = cond ? tmp : 0.0f;
            float v2 = v * v;
            s += v2;
            cnt += (v2 > 0.0f) ? 1 : 0;
        }
    }
    __shared__ float ss[256];
    __shared__ int   sc[256];
    ss[t] = s; sc[t] = cnt;
    __syncthreads();
    for (int o = 128; o > 0; o >>= 1) {
        if (t < o) { ss[t] += ss[t + o]; sc[t] += sc[t + o]; }
        __syncthreads();
    }
    if (t == 0) {
        float times = fmaxf((float)sc[0], 1.0f);
        float nm = ss[0] / times;
        float tm = (tg - MARGIN_F) - (1.0f + tg) * nm;
        out[(size_t)n * C_COLS + lab] = tm * SCALE_F;
    }
}

extern "C" void kernel_launch(void* const* d_in, const int* in_sizes, int n_in,
                              void* d_out, int out_size, void* d_ws, size_t ws_size,
                              hipStream_t stream) {
    const float* emb   = (const float*)d_in[0];
    const float* K     = (const float*)d_in[1];
    const int*   label = (const int*)d_in[2];
    float* out = (float*)d_out;

    float* ws       = (float*)d_ws;
    float* normA    = ws + WS_NORMA;
    float* tgt      = ws + WS_TGT;
    unsigned* scal  = (unsigned*)(ws + WS_SCAL);
    unsigned* h1    = (unsigned*)(ws + WS_H1);
    unsigned* h2    = (unsigned*)(ws + WS_H2);
    unsigned* h3    = (unsigned*)(ws + WS_H3);

    const dim3 histGrid(13, N_ROWS);

    ws_init_kernel<<<34, 256, 0, stream>>>(scal);
    rownorm_kernel<<<N_ROWS, 256, 0, stream>>>(emb, normA);
    gemm_cos_kernel<<<(C_COLS + 63) / 64, 256, 0, stream>>>(normA, K, out);
    tgt_kernel<<<1, 256, 0, stream>>>(out, label, tgt);
    hist1_kernel<<<histGrid, 256, 0, stream>>>(out, label, tgt, scal, h1);
    scan1_kernel<<<1, 32, 0, stream>>>(scal, h1);
    hist2_kernel<<<histGrid, 256, 0, stream>>>(out, label, tgt, scal, h2);
    scan2_kernel<<<1, 32, 0, stream>>>(scal, h2);
    hist3_kernel<<<histGrid, 256, 0, stream>>>(out, label, tgt, scal, h3);
    scan3_kernel<<<1, 32, 0, stream>>>(scal, h3);
    negfix_kernel<<<N_ROWS, 256, 0, stream>>>(out, label, tgt, scal);
}